// QNN_72988674228630
// MI455X (gfx1250) — compile-verified
//
#include <hip/hip_runtime.h>

// CDNA5 / gfx1250 (MI455X), wave32. Full-fp32 quantum circuit simulation via
// rank-factorized state evolution; dense 32x32x32 matmuls on the rank factors
// (batched over 32 circuit instances per wave) run on v_wmma_f32_16x16x4_f32.

typedef __attribute__((ext_vector_type(2))) float v2f;
typedef __attribute__((ext_vector_type(8))) float v8f;

#define NQ   10
#define NB   128
#define NF   64
#define UST  33   // padded LDS stride (floats) for 32x32 matrices -> no bank conflicts
#define XST  33

// Forward map of the 5-bit CNOT chain (b4->b3),(b3->b2),(b2->b1),(b1->b0):
// amplitude at index m moves to cnot5_fwd(m).
__device__ __forceinline__ int cnot5_fwd(int m) {
  m ^= ((m >> 4) & 1) << 3;
  m ^= ((m >> 3) & 1) << 2;
  m ^= ((m >> 2) & 1) << 1;
  m ^= ((m >> 1) & 1);
  return m;
}
// Inverse (CNOTs are involutions; apply in reverse order).
__device__ __forceinline__ int cnot5_inv(int m) {
  m ^= ((m >> 1) & 1);
  m ^= ((m >> 2) & 1) << 1;
  m ^= ((m >> 3) & 1) << 2;
  m ^= ((m >> 4) & 1) << 3;
  return m;
}

// A-operand fragments of a 32x32 LDS matrix for V_WMMA_F32_16X16X4_F32.
// ISA layout (16x4 f32 A): lanes 0-15: M=lane, VGPR0/1 = K0/K1;
// lanes 16-31: M=lane-16, VGPR0/1 = K2/K3.
__device__ __forceinline__ void load_a_frags(const float* U, int lane, v2f af[2][8]) {
  const int half = lane >> 4, r = lane & 15;
#pragma unroll
  for (int i = 0; i < 2; ++i)
#pragma unroll
    for (int kb = 0; kb < 8; ++kb) {
      const int row = i * 16 + r;
      const int k0  = kb * 4 + half * 2;
      v2f a;
      a.x = U[row * UST + k0];
      a.y = U[row * UST + k0 + 1];
      af[i][kb] = a;
    }
}

// D = Mat(32x32, A-frags) x X(32x32 in LDS, row-major stride XST).
// B-operand (4x16 f32): VGPR0: lanes0-15 K=0 / lanes16-31 K=2; VGPR1: K=1 / K=3
// => per lane component x is an EVEN row of X, y an ODD row.
// MODE: 0 plain; 1 even rows only (E mask); 2 odd rows only (O mask);
//       3 row k -> cnot5_inv(k)        (reads Pc*beta)
//       4 row k -> cnot5_inv(k) ^ 16   (reads Pc*F*beta, F = flip col MSB)
template <int MODE>
__device__ __forceinline__ void mm32(const v2f af[2][8], const float* X, int lane,
                                     v8f D[2][2]) {
  const int half = lane >> 4, nlo = lane & 15;
#pragma unroll
  for (int i = 0; i < 2; ++i)
#pragma unroll
    for (int j = 0; j < 2; ++j)
      D[i][j] = {};
#pragma unroll
  for (int kb = 0; kb < 8; ++kb) {
    const int k0 = kb * 4 + half * 2;  // always even
    const int k1 = k0 + 1;             // always odd
#pragma unroll
    for (int j = 0; j < 2; ++j) {
      const int n = j * 16 + nlo;
      v2f b;
      if (MODE == 0)      { b.x = X[k0 * XST + n];            b.y = X[k1 * XST + n]; }
      else if (MODE == 1) { b.x = X[k0 * XST + n];            b.y = 0.0f; }
      else if (MODE == 2) { b.x = 0.0f;                        b.y = X[k1 * XST + n]; }
      else if (MODE == 3) { b.x = X[cnot5_inv(k0) * XST + n];  b.y = X[cnot5_inv(k1) * XST + n]; }
      else                { b.x = X[(cnot5_inv(k0) ^ 16) * XST + n];
                            b.y = X[(cnot5_inv(k1) ^ 16) * XST + n]; }
#pragma unroll
      for (int i = 0; i < 2; ++i)
        D[i][j] = __builtin_amdgcn_wmma_f32_16x16x4_f32(
            false, af[i][kb], false, b, (short)0, D[i][j], false, false);
    }
  }
}

// Store D tiles (C/D layout: VGPR v: lanes0-15 -> M=v, lanes16-31 -> M=v+8)
// to LDS, optionally applying the CNOT row permutation during the store.
template <bool PERM>
__device__ __forceinline__ void store_d(float* X, int lane, const v8f D[2][2]) {
  const int half = lane >> 4, nlo = lane & 15;
#pragma unroll
  for (int i = 0; i < 2; ++i)
#pragma unroll
    for (int j = 0; j < 2; ++j)
#pragma unroll
      for (int v = 0; v < 8; ++v) {
        const int m   = i * 16 + v + 8 * half;
        const int row = PERM ? cnot5_fwd(m) : m;
        X[row * XST + j * 16 + nlo] = D[i][j][v];
      }
}

__global__ __launch_bounds__(32) void qnn_state_wmma(const float* __restrict__ x,
                                                     const float* __restrict__ w,
                                                     float* __restrict__ out) {
  __shared__ float Umat[4][32 * UST];  // U1R, U1C, U2R, U2C (16.9 KB)
  __shared__ float Xbuf[4][32 * XST];  // A0, B0, alpha, beta (16.9 KB)

  const int lane  = threadIdx.x;            // one wave32 per workgroup
  const int f     = blockIdx.x >> 2;        // out_feature
  const int bbase = (blockIdx.x & 3) * 32;  // batch group base

  // ---- Stage 1: build the four 32x32 Kronecker RY matrices (per f) ----
  for (int idx = lane; idx < 128; idx += 32) {
    const int mm    = idx >> 5;        // 0:U1R 1:U1C 2:U2R 3:U2C
    const int j     = idx & 31;
    const int layer = mm >> 1;
    const int q0    = (mm & 1) * 5;    // row side: qubits 0-4, col side: 5-9
    float c[5], s[5];
#pragma unroll
    for (int q = 0; q < 5; ++q) {
      const float h = 0.5f * w[(size_t)f * NQ * 2 + (size_t)(q0 + q) * 2 + layer];
      c[q] = cosf(h);
      s[q] = sinf(h);
    }
    for (int r = 0; r < 32; ++r) {
      float p = 1.0f;
#pragma unroll
      for (int q = 0; q < 5; ++q) {
        const int rb = (r >> (4 - q)) & 1;
        const int jb = (j >> (4 - q)) & 1;
        p *= (rb == jb) ? c[q] : ((rb == 1) ? s[q] : -s[q]);  // RY=[[c,-s],[s,c]]
      }
      Umat[mm][r * UST + j] = p;
    }
  }

  // ---- Stage 2: encoding product-state vectors; lane = batch column ----
  {
    const int batch = bbase + lane;
    float cs[10], sn[10];
#pragma unroll
    for (int q = 0; q < NQ; ++q) {
      const float h = 0.5f * x[batch * NQ + q];
      cs[q] = cosf(h);
      sn[q] = sinf(h);
    }
    float arr[32];
#pragma unroll
    for (int side = 0; side < 2; ++side) {   // 0: rows (q0-4), 1: cols (q5-9)
      arr[0] = 1.0f;
      int len = 1;
      for (int q = 0; q < 5; ++q) {          // Kronecker doubling, q appended as LSB
        const float cq = cs[side * 5 + q], sq = sn[side * 5 + q];
        for (int i2 = len - 1; i2 >= 0; --i2) {
          const float vv = arr[i2];
          arr[2 * i2]     = cq * vv;
          arr[2 * i2 + 1] = sq * vv;
        }
        len <<= 1;
      }
      for (int r = 0; r < 32; ++r) Xbuf[side][r * XST + lane] = arr[r];
    }
  }
  __syncthreads();

  // ---- Stage 3: layer-1 rotations (WMMA), CNOT perms folded into stores ----
  v2f af[2][8];
  v8f D[2][2];
  load_a_frags(Umat[0], lane, af);
  mm32<0>(af, Xbuf[0], lane, D);
  store_d<true>(Xbuf[2], lane, D);           // alpha = P1R * U1R * A0
  load_a_frags(Umat[1], lane, af);
  mm32<0>(af, Xbuf[1], lane, D);
  store_d<false>(Xbuf[3], lane, D);          // beta  = U1C * B0
  __syncthreads();

  // ---- Stage 4: layer-2 rotations on the rank factors (stay in registers) ----
  v8f u1[2][2], u2[2][2], r1[2][2], r2[2][2];
  load_a_frags(Umat[2], lane, af);
  mm32<1>(af, Xbuf[2], lane, u1);            // U2R * (E alpha)
  mm32<2>(af, Xbuf[2], lane, u2);            // U2R * (O alpha)
  load_a_frags(Umat[3], lane, af);
  mm32<3>(af, Xbuf[3], lane, r1);            // U2C * (Pc beta)
  mm32<4>(af, Xbuf[3], lane, r2);            // U2C * (Pc F beta)

  // ---- Stage 5: Gram reductions; layer-2 CNOTs folded into weights ----
  // Left pairs (row perm P applied as weight index fm=P(m); parity = crossing split):
  //  accL: [0]d11 [1]dz11 [2]d22 [3]dz22 [4]d33 [5]dz33 [6]d44 [7]dz44 [8]d13 [9]dz13 [10]d24 [11]dz24
  // Right: accR: [0]e11 [1]ezA11 [2]ezB11 [3]e33 [4]ezA33 [5]ezB33 [6]e13 [7]ezA13 [8]ezB13
  float accL[2][12], accR[2][9];
#pragma unroll
  for (int j = 0; j < 2; ++j) {
#pragma unroll
    for (int t = 0; t < 12; ++t) accL[j][t] = 0.0f;
#pragma unroll
    for (int t = 0; t < 9; ++t)  accR[j][t] = 0.0f;
  }
  const int half = lane >> 4;
#pragma unroll
  for (int i = 0; i < 2; ++i)
#pragma unroll
    for (int v = 0; v < 8; ++v) {
      const int   m   = i * 16 + v + 8 * half;
      const int   fm  = cnot5_fwd(m);
      const float zrw = 5.0f - 2.0f * (float)__popc(fm);              // Z-diag, row half
      const float zA  = zrw;                                          // zc[f(m)]
      const float zB  = 5.0f - 2.0f * (float)__popc(cnot5_fwd(m ^ 16)); // zc[f(m^16)]
      const bool  odd = (fm & 1) != 0;
#pragma unroll
      for (int j = 0; j < 2; ++j) {
        const float a1 = u1[i][j][v], a2 = u2[i][j][v];
        const float q1 = a1 * a1, q2 = a2 * a2, q3 = a1 * a2;
        if (!odd) {
          accL[j][0] += q1; accL[j][1] += q1 * zrw;
          accL[j][4] += q2; accL[j][5] += q2 * zrw;
          accL[j][8] += q3; accL[j][9] += q3 * zrw;
        } else {
          accL[j][2]  += q1; accL[j][3]  += q1 * zrw;
          accL[j][6]  += q2; accL[j][7]  += q2 * zrw;
          accL[j][10] += q3; accL[j][11] += q3 * zrw;
        }
        const float b1 = r1[i][j][v], b2 = r2[i][j][v];
        const float p1 = b1 * b1, p2 = b2 * b2, p3 = b1 * b2;
        accR[j][0] += p1; accR[j][1] += p1 * zA; accR[j][2] += p1 * zB;
        accR[j][3] += p2; accR[j][4] += p2 * zA; accR[j][5] += p2 * zB;
        accR[j][6] += p3; accR[j][7] += p3 * zA; accR[j][8] += p3 * zB;
      }
    }
  // Columns are split across lane pairs (L, L^16): one cross-half reduction.
#pragma unroll
  for (int j = 0; j < 2; ++j) {
#pragma unroll
    for (int t = 0; t < 12; ++t) accL[j][t] += __shfl_xor(accL[j][t], 16);
#pragma unroll
    for (int t = 0; t < 9; ++t)  accR[j][t] += __shfl_xor(accR[j][t], 16);
  }

  if (lane < 16) {
#pragma unroll
    for (int j = 0; j < 2; ++j) {
      const float o =
          accL[j][1] * accR[j][0] + accL[j][0] * accR[j][1]            // (1,1)
        + accL[j][3] * accR[j][0] + accL[j][2] * accR[j][2]            // (2,2)
        + accL[j][5] * accR[j][3] + accL[j][4] * accR[j][4]            // (3,3)
        + accL[j][7] * accR[j][3] + accL[j][6] * accR[j][5]            // (4,4)
        + 2.0f * (accL[j][9]  * accR[j][6] + accL[j][8]  * accR[j][7]) // (1,3)
        + 2.0f * (accL[j][11] * accR[j][6] + accL[j][10] * accR[j][8]);// (2,4)
      out[f * NB + bbase + j * 16 + lane] = o;
    }
  }
}

extern "C" void kernel_launch(void* const* d_in, const int* in_sizes, int n_in,
                              void* d_out, int out_size, void* d_ws, size_t ws_size,
                              hipStream_t stream) {
  const float* x = (const float*)d_in[0];   // (128, 10) f32
  const float* w = (const float*)d_in[1];   // (64, 10, 2) f32
  float* out = (float*)d_out;               // (64, 128) f32
  (void)in_sizes; (void)n_in; (void)out_size; (void)d_ws; (void)ws_size;
  qnn_state_wmma<<<dim3(NF * 4), dim3(32), 0, stream>>>(x, w, out);
}